// CorrLookup_5308579577971
// MI455X (gfx1250) — compile-verified
//
#include <hip/hip_runtime.h>
#include <hip/hip_bf16.h>

// RAFT-style multi-level correlation lookup, specialized for
// B=2, H=64, W=128, radius=4 (pyramid levels 0..3).
//
// MI455X (wave32) strategy:
//  - thread = (pixel n, level); 65536 threads = 2048 waves (latency hiding).
//  - All 81 taps share one fractional weight pair (wx,wy), so load a 10x10
//    patch once and interpolate separably, streaming row-by-row (two 9-wide
//    rows of intermediates live -> low VGPR pressure).
//  - Compile-time dims: all index math is shifts/masks, all 81 store offsets
//    are 24-bit immediate offsets off one base address.
//  - Interior fast path: unclamped immediate-offset loads; border path keeps
//    branch-free clamp + cndmask, bit-faithful to clip + where(valid,v,0).
//  - Non-temporal loads (corr is single-use streaming; corr0=512MB > L2) and
//    non-temporal stores (write-once output).

#define RADIUS 4
#define KW 9      // window width  (2*RADIUS+1)
#define PW 10     // patch width   (KW+1)

template <int LVL>
__device__ __forceinline__ void corr_level(const float* __restrict__ corr,
                                           const float* __restrict__ flow,
                                           float* __restrict__ out,
                                           unsigned int n)
{
    constexpr int H  = 64;
    constexpr int W  = 128;
    constexpr int HW = H * W;                 // 8192
    constexpr int Hc = H >> LVL;
    constexpr int Wc = W >> LVL;
    constexpr float scale = 1.0f / (float)(1 << LVL);   // exact

    const unsigned int b  = n >> 13;          // n / 8192
    const unsigned int hw = n & (HW - 1);
    const unsigned int h  = hw >> 7;          // hw / 128
    const unsigned int w  = hw & (W - 1);

    // flow (B, 2, H, W): ch0 = x displacement, ch1 = y.
    const float fx = flow[b * 2 * HW + hw];
    const float fy = flow[b * 2 * HW + HW + hw];

    const float cx = ((float)w + fx) * scale;
    const float cy = ((float)h + fy) * scale;

    const float fx0 = floorf(cx);
    const float fy0 = floorf(cy);
    const float wx  = cx - fx0;               // shared by all 81 taps
    const float wy  = cy - fy0;
    const int   x0  = (int)fx0;
    const int   y0  = (int)fy0;
    const float omwx = 1.0f - wx;
    const float omwy = 1.0f - wy;

    const float* __restrict__ slab = corr + (size_t)n * (Hc * Wc);

    // out (B, 4*81, H, W); channel = LVL*81 + k, k = iy*9 + ix,
    // tap k samples at (x = cx + (iy-4), y = cy + (ix-4)).
    float* __restrict__ outBase = out + (size_t)b * (4 * KW * KW * HW)
                                      + (size_t)(LVL * KW * KW) * HW + hw;

    const bool interior = (x0 >= RADIUS) && (x0 + RADIUS + 1 <= Wc - 1) &&
                          (y0 >= RADIUS) && (y0 + RADIUS + 1 <= Hc - 1);

    float prev[KW];
    float cur[KW];

    #pragma unroll
    for (int r = 0; r < PW; ++r) {
        float tmp[PW];
        if (interior) {
            // Unclamped: base + compile-time offsets -> immediate-offset NT loads.
            const float* __restrict__ row =
                slab + (size_t)(y0 - RADIUS + r) * Wc + (x0 - RADIUS);
            #pragma unroll
            for (int c = 0; c < PW; ++c)
                tmp[c] = __builtin_nontemporal_load(row + c);
        } else {
            const int  yi = y0 - RADIUS + r;
            const bool rowValid = (yi >= 0) && (yi < Hc);
            const int  yc = min(max(yi, 0), Hc - 1);
            const float* __restrict__ row = slab + (size_t)yc * Wc;
            #pragma unroll
            for (int c = 0; c < PW; ++c) {
                const int  xi = x0 - RADIUS + c;
                const int  xc = min(max(xi, 0), Wc - 1);
                const bool v  = rowValid && (xi >= 0) && (xi < Wc);
                const float val = __builtin_nontemporal_load(row + xc);
                tmp[c] = v ? val : 0.0f;
            }
        }

        // Interpolate along x within this row.
        #pragma unroll
        for (int iy = 0; iy < KW; ++iy)
            cur[iy] = omwx * tmp[iy] + wx * tmp[iy + 1];

        // Combine with previous row along y -> emit column ix = r-1.
        if (r > 0) {
            const int ix = r - 1;
            #pragma unroll
            for (int iy = 0; iy < KW; ++iy) {
                const float val = omwy * prev[iy] + wy * cur[iy];
                __builtin_nontemporal_store(val, outBase + (iy * KW + ix) * HW);
            }
        }
        #pragma unroll
        for (int iy = 0; iy < KW; ++iy) prev[iy] = cur[iy];
    }
}

__global__ __launch_bounds__(256) void corr_lookup_kernel(
    const float* __restrict__ c0, const float* __restrict__ c1,
    const float* __restrict__ c2, const float* __restrict__ c3,
    const float* __restrict__ flow, float* __restrict__ out)
{
    // Grid exactly covers N = 16384 pixels (64 blocks x 256 threads).
    const unsigned int n = blockIdx.x * 256u + threadIdx.x;
    switch (blockIdx.y) {                     // uniform scalar branch
        case 0: corr_level<0>(c0, flow, out, n); break;
        case 1: corr_level<1>(c1, flow, out, n); break;
        case 2: corr_level<2>(c2, flow, out, n); break;
        default: corr_level<3>(c3, flow, out, n); break;
    }
}

extern "C" void kernel_launch(void* const* d_in, const int* in_sizes, int n_in,
                              void* d_out, int out_size, void* d_ws, size_t ws_size,
                              hipStream_t stream) {
    (void)in_sizes; (void)n_in; (void)d_ws; (void)ws_size; (void)out_size;

    const float* c0   = (const float*)d_in[0];
    const float* c1   = (const float*)d_in[1];
    const float* c2   = (const float*)d_in[2];
    const float* c3   = (const float*)d_in[3];
    const float* flow = (const float*)d_in[4];
    float* out        = (float*)d_out;

    dim3 block(256, 1, 1);
    dim3 grid(64, 4, 1);                      // 16384 pixels x 4 levels
    corr_lookup_kernel<<<grid, block, 0, stream>>>(c0, c1, c2, c3, flow, out);
}